// RNN_Encoder_1563368095997
// MI455X (gfx1250) — compile-verified
//
#include <hip/hip_runtime.h>

// Problem constants (match reference)
constexpr int B = 1024;
constexpr int T = 512;
constexpr int V = 32000;
constexpr int E = 32;
constexpr int H = 16;

typedef __attribute__((ext_vector_type(2))) float v2f;
typedef __attribute__((ext_vector_type(4))) float v4f;
typedef __attribute__((ext_vector_type(8))) float v8f;
typedef __attribute__((ext_vector_type(4))) int   v4i;

// ---------------------------------------------------------------------------
// Phase 1: embW[v,h] = sum_e emb[v,e]*W[e,h] + b[h]   (2 MB table, L2-resident)
// Folds embedding gather + input GEMM + bias into a single lookup table.
// ---------------------------------------------------------------------------
__global__ void embw_precompute(const float* __restrict__ emb,
                                const float* __restrict__ W,
                                const float* __restrict__ b,
                                float* __restrict__ embW) {
    int idx = blockIdx.x * blockDim.x + threadIdx.x;   // 0 .. V*H-1
    if (idx >= V * H) return;
    int v = idx >> 4;
    int h = idx & 15;
    float s = b[h];
    const float* er = emb + v * E;
#pragma unroll
    for (int e = 0; e < E; ++e)
        s = fmaf(er[e], W[e * H + h], s);
    embW[idx] = s;
}

// Native CDNA5 transcendental v_tanh_f32 (confirmed lowering).
__device__ __forceinline__ float tanh_fast(float x) {
#if __has_builtin(__builtin_amdgcn_tanhf)
    return __builtin_amdgcn_tanhf(x);
#else
    float e = __builtin_amdgcn_exp2f(x * 2.885390081777926f);  // 2*log2(e)
    return fmaf(-2.0f, __builtin_amdgcn_rcpf(e + 1.0f), 1.0f);
#endif
}

// Half-wave swap (lane i <-> lane i^16): raw v_permlanex16_b32 so the dest
// is not register-tied (the builtin's `old` operand forced a v_mov per use).
// One lane-select in an SGPR, the other as the single allowed 32-bit literal
// (same operand pattern the compiler emits). EXEC is all-ones here; fi/bc=0.
__device__ __forceinline__ float xswap16(float x) {
    float r;
    asm("v_permlanex16_b32 %0, %1, %2, 0xfedcba98"
        : "=v"(r)
        : "v"(x), "s"(0x76543210));
    return r;
}

// ---------------------------------------------------------------------------
// Phase 2: recurrence. One wave owns TWO independent 16-row batch tiles
// (interleaved dependency chains fill each other's stalls). State kept
// transposed: g = h^T, step: g = tanh( x^T + U^T * g ) with
// V_WMMA_F32_16X16X4_F32, K=16 as 4 chunks over two accumulator chains.
// x-tiles software-pipelined 4 deep (tokens fetched 8 ahead as b128 quads,
// embW row gathers issued 4 steps ahead -> L2 latency fully hidden).
// All global accesses use uniform SGPR base + unsigned 32-bit offsets so the
// backend emits the saddr+voffset global form with folded immediate offsets.
//
// Layouts (per CDNA5 ISA):
//   C/D 16x16 f32: vgpr j, lane n<16 -> (M=j, N=n); n>=16 -> (M=j+8, N=n-16)
//   A  16x4  f32: vgpr0: K=0 (lanes 0-15), K=2 (lanes 16-31); vgpr1: K=1,K=3
//   B   4x16 f32 (mirror of A). Feedback C->B = half-wave swap + select.
// ---------------------------------------------------------------------------
__global__ __launch_bounds__(32) void rnn_scan(const int* __restrict__ X,
                                               const float* __restrict__ U,
                                               const float* __restrict__ embW,
                                               float* __restrict__ out) {
    const int lane = threadIdx.x;           // 0..31, wave32
    const int base = blockIdx.x * 32;       // two 16-row tiles per wave
    const int col  = lane & 15;             // batch column within tile
    const bool lo  = lane < 16;
    const uint32_t hoff = lo ? 0u : 8u;     // h-range this lane-half covers

    // ---- A = U^T in WMMA A-layout, loop-invariant, shared by both tiles ----
    const int koff = lo ? 0 : 2;
    v2f a0, a1, a2, a3;
    a0[0] = U[(0 + koff) * H + col];  a0[1] = U[(1 + koff) * H + col];
    a1[0] = U[(4 + koff) * H + col];  a1[1] = U[(5 + koff) * H + col];
    a2[0] = U[(8 + koff) * H + col];  a2[1] = U[(9 + koff) * H + col];
    a3[0] = U[(12 + koff) * H + col]; a3[1] = U[(13 + koff) * H + col];

    // 32-bit element offsets from uniform bases
    const uint32_t xoff0 = (uint32_t)(base + col) * T;          // into X
    const uint32_t xoff1 = xoff0 + 16u * T;
    const uint32_t ooff0 = (uint32_t)(base + col) * T * H + hoff; // into out
    const uint32_t ooff1 = ooff0 + 16u * T * H;

    v8f g0 = {}, g1 = {};

    auto step_tile = [&](v8f& g, v4f& e0s, v4f& e1s, int ntok, float* op) {
        // x tile (C layout) seeds accumulator chain 0 -> "+x" is free
        v8f acc0 = {e0s[0], e0s[1], e0s[2], e0s[3],
                    e1s[0], e1s[1], e1s[2], e1s[3]};
        // refill this pipeline slot: gather tile for t+4 (saddr + 32-bit off)
        const float* np = embW + ((uint32_t)ntok * 16u + hoff);
        e0s = *reinterpret_cast<const v4f*>(np);
        e1s = *reinterpret_cast<const v4f*>(np + 4);

        // C-layout -> B-layout feedback: half-wave swap + select
        float hs0 = xswap16(g[0]), hs1 = xswap16(g[1]);
        float hs2 = xswap16(g[2]), hs3 = xswap16(g[3]);
        float hs4 = xswap16(g[4]), hs5 = xswap16(g[5]);
        float hs6 = xswap16(g[6]), hs7 = xswap16(g[7]);
        v2f b0 = {lo ? g[0] : hs2, lo ? g[1] : hs3};   // K rows 0..3
        v2f b1 = {lo ? g[4] : hs6, lo ? g[5] : hs7};   // K rows 4..7
        v2f b2 = {lo ? hs0 : g[2], lo ? hs1 : g[3]};   // K rows 8..11
        v2f b3 = {lo ? hs4 : g[6], lo ? hs5 : g[7]};   // K rows 12..15

        // U^T * g : two independent 2-deep WMMA chains
        v8f acc1 = {};
        acc0 = __builtin_amdgcn_wmma_f32_16x16x4_f32(false, a0, false, b0,
                                                     (short)0, acc0, false, false);
        acc1 = __builtin_amdgcn_wmma_f32_16x16x4_f32(false, a2, false, b2,
                                                     (short)0, acc1, false, false);
        acc0 = __builtin_amdgcn_wmma_f32_16x16x4_f32(false, a1, false, b1,
                                                     (short)0, acc0, false, false);
        acc1 = __builtin_amdgcn_wmma_f32_16x16x4_f32(false, a3, false, b3,
                                                     (short)0, acc1, false, false);
        v8f s = acc0 + acc1;
#pragma unroll
        for (int j = 0; j < 8; ++j) g[j] = tanh_fast(s[j]);

        // coalesced store: h contiguous -> two b128 per lane
        v4f s0 = {g[0], g[1], g[2], g[3]};
        v4f s1 = {g[4], g[5], g[6], g[7]};
        *reinterpret_cast<v4f*>(op) = s0;
        *reinterpret_cast<v4f*>(op + 4) = s1;
    };

    // ---- prologue: tokens 0..7 as two b128 quads, issue tile loads 0..3 ----
    v4i q0_0 = *reinterpret_cast<const v4i*>(X + xoff0);
    v4i q0_1 = *reinterpret_cast<const v4i*>(X + xoff1);
    v4i tkA0 = *reinterpret_cast<const v4i*>(X + (xoff0 + 4u));
    v4i tkA1 = *reinterpret_cast<const v4i*>(X + (xoff1 + 4u));

    v4f pe0_0[4], pe1_0[4], pe0_1[4], pe1_1[4];
#pragma unroll
    for (int i = 0; i < 4; ++i) {
        const float* p0 = embW + ((uint32_t)q0_0[i] * 16u + hoff);
        pe0_0[i] = *reinterpret_cast<const v4f*>(p0);
        pe1_0[i] = *reinterpret_cast<const v4f*>(p0 + 4);
        const float* p1 = embW + ((uint32_t)q0_1[i] * 16u + hoff);
        pe0_1[i] = *reinterpret_cast<const v4f*>(p1);
        pe1_1[i] = *reinterpret_cast<const v4f*>(p1 + 4);
    }

    // ---- main loop: 4-step blocks, 4-deep tile pipeline, 8-ahead tokens ----
    for (int tt = 0; tt < T; tt += 4) {
        int tq = (tt + 8 < T) ? (tt + 8) : (T - 4);   // clamped, 16B aligned
        v4i tkB0 = *reinterpret_cast<const v4i*>(X + (xoff0 + (uint32_t)tq));
        v4i tkB1 = *reinterpret_cast<const v4i*>(X + (xoff1 + (uint32_t)tq));
        float* opb0 = out + (ooff0 + (uint32_t)tt * H);  // u*64B folds to imm
        float* opb1 = out + (ooff1 + (uint32_t)tt * H);
#pragma unroll
        for (int u = 0; u < 4; ++u) {
            step_tile(g0, pe0_0[u], pe1_0[u], tkA0[u], opb0 + u * H);
            step_tile(g1, pe0_1[u], pe1_1[u], tkA1[u], opb1 + u * H);
        }
        tkA0 = tkB0;
        tkA1 = tkB1;
    }

    // ---- final hidden states hT for both tiles ----
    float* hp0 = out + ((uint32_t)B * T * H + (uint32_t)(base + col) * H + hoff);
    float* hp1 = hp0 + 16 * H;
    v4f f00 = {g0[0], g0[1], g0[2], g0[3]};
    v4f f01 = {g0[4], g0[5], g0[6], g0[7]};
    *reinterpret_cast<v4f*>(hp0) = f00;
    *reinterpret_cast<v4f*>(hp0 + 4) = f01;
    v4f f10 = {g1[0], g1[1], g1[2], g1[3]};
    v4f f11 = {g1[4], g1[5], g1[6], g1[7]};
    *reinterpret_cast<v4f*>(hp1) = f10;
    *reinterpret_cast<v4f*>(hp1 + 4) = f11;
}

extern "C" void kernel_launch(void* const* d_in, const int* in_sizes, int n_in,
                              void* d_out, int out_size, void* d_ws, size_t ws_size,
                              hipStream_t stream) {
    const int*   X   = (const int*)d_in[0];
    const float* emb = (const float*)d_in[1];
    const float* W   = (const float*)d_in[2];
    const float* U   = (const float*)d_in[3];
    const float* b   = (const float*)d_in[4];
    float* out  = (float*)d_out;
    float* embW = (float*)d_ws;    // V*H*4 = 2 MB scratch, L2-resident

    // Phase 1: fold emb@W + b into a gather table
    embw_precompute<<<(V * H + 255) / 256, 256, 0, stream>>>(emb, W, b, embW);

    // Phase 2: 32 waves, each runs the T=512 scan for two 16-row batch tiles
    rnn_scan<<<B / 32, 32, 0, stream>>>(X, U, embW, out);
}